// SurrogateAUCLossDynamicWeighted_4406636445810
// MI455X (gfx1250) — compile-verified
//
#include <hip/hip_runtime.h>

typedef __attribute__((ext_vector_type(2))) float v2f;
typedef __attribute__((ext_vector_type(8))) float v8f;

#define BN 512
#define LN 16

// ---------------------------------------------------------------------------
// Kernel 1: S[i][j] = sum_{c,d} relu(1 - yp[i][c] + yp[j][d])
// One 256-thread block computes a 16x16 tile of S; the 16 i-rows and 16
// j-rows of yp (16 floats each) are staged in LDS once per block.
// ---------------------------------------------------------------------------
__global__ void __launch_bounds__(256)
k_pairwise_hinge(const float* __restrict__ yp, float* __restrict__ S) {
    __shared__ float As[16][16];
    __shared__ float Bs[16][16];
    const int t  = threadIdx.x;       // 0..255
    const int ti = t >> 4;
    const int tj = t & 15;
    const int i0 = blockIdx.y * 16;
    const int j0 = blockIdx.x * 16;
    As[ti][tj] = yp[(i0 + ti) * LN + tj];
    Bs[ti][tj] = yp[(j0 + ti) * LN + tj];
    __syncthreads();
    float acc = 0.0f;
#pragma unroll
    for (int c = 0; c < 16; ++c) {
        const float a = 1.0f - As[ti][c];
#pragma unroll
        for (int d = 0; d < 16; ++d) {
            acc += fmaxf(a + Bs[tj][d], 0.0f);
        }
    }
    S[(i0 + ti) * BN + (j0 + tj)] = acc;
}

// ---------------------------------------------------------------------------
// Kernel 2: U = mp^T * S  via V_WMMA_F32_16X16X4_F32.
//   U[l][j] = sum_k mp[k][l] * S[k][j],  mp[k][l] = (y_true[k][l]==1)
// One wave (32 threads) per 16-column stripe of S; K=512 consumed in steps
// of 4 -> 128 chained WMMAs accumulating into one 16x16 f32 tile. Exact in
// f32 (masks are 0/1). S is read exactly once across all 32 waves.
// ---------------------------------------------------------------------------
__global__ void __launch_bounds__(32)
k_mask_gemm(const int* __restrict__ yt, const float* __restrict__ S,
            float* __restrict__ U) {
    const int t  = threadIdx.x;   // 0..31
    const int m  = t & 15;        // A: row M / B: col N within tile
    const int hi = t >> 4;        // 0: K={0,1}, 1: K={2,3}
    const int n  = blockIdx.x * 16 + m;

    v8f c = {};
    for (int k0 = 0; k0 < BN; k0 += 4) {
        const int ka = k0 + 2 * hi;
        v2f a, b;
        // A-matrix 16x4 f32 layout: lane m, VGPR0=K(ka), VGPR1=K(ka+1)
        a.x = (yt[ka * LN + m]       == 1) ? 1.0f : 0.0f;
        a.y = (yt[(ka + 1) * LN + m] == 1) ? 1.0f : 0.0f;
        // B-matrix 4x16 f32 layout: lane n, VGPR0=K(ka), VGPR1=K(ka+1)
        b.x = S[ka * BN + n];
        b.y = S[(ka + 1) * BN + n];
        // (neg_a, A, neg_b, B, c_mod, C, reuse_a, reuse_b)
        c = __builtin_amdgcn_wmma_f32_16x16x4_f32(
                false, a, false, b, (short)0, c, false, false);
    }
    // C/D layout: lane t holds rows g + 8*hi, col n
#pragma unroll
    for (int g = 0; g < 8; ++g) {
        U[(g + 8 * hi) * BN + n] = c[g];
    }
}

// ---------------------------------------------------------------------------
// Kernel 3: deterministic finalize.
//   per_label[l] = sum_j U[l][j] * (y_true[j][l]!=1)
//   n_pos[l]     = sum_j (y_true[j][l]==1),  n_neg = 512 - n_pos
//   loss = sum_l valid ? per_label / ((n_pos*n_neg)^2 * 256) : 0
// 256 threads: 16 labels x 16 partial-reducers (32 j's each), then an
// in-order tree finish. No float atomics -> bitwise-deterministic.
// ---------------------------------------------------------------------------
__global__ void __launch_bounds__(256)
k_finalize(const int* __restrict__ yt, const float* __restrict__ U,
           float* __restrict__ out) {
    __shared__ float pl[16][16];
    __shared__ int   pc[16][16];
    const int t    = threadIdx.x;
    const int l    = t >> 4;
    const int part = t & 15;

    float s = 0.0f;
    int np = 0;
    const int jb = part * 32;
    for (int j = jb; j < jb + 32; ++j) {
        const int ispos = (yt[j * LN + l] == 1);
        np += ispos;
        s += ispos ? 0.0f : U[l * BN + j];
    }
    pl[l][part] = s;
    pc[l][part] = np;
    __syncthreads();

    if (t < 16) {                  // t = label index
        float per = 0.0f;
        int npos = 0;
        for (int p = 0; p < 16; ++p) { per += pl[t][p]; npos += pc[t][p]; }
        const int nneg = BN - npos;
        float val = 0.0f;
        if (npos > 0 && nneg > 0) {
            float d = (float)npos * (float)nneg;   // match f32 reference math
            d = d * d * (float)(LN * LN);
            val = per / d;
        }
        pl[t][0] = val;
    }
    __syncthreads();

    if (t == 0) {
        float loss = 0.0f;
        for (int q = 0; q < 16; ++q) loss += pl[q][0];
        out[0] = loss;
    }
}

// ---------------------------------------------------------------------------
extern "C" void kernel_launch(void* const* d_in, const int* in_sizes, int n_in,
                              void* d_out, int out_size, void* d_ws, size_t ws_size,
                              hipStream_t stream) {
    (void)in_sizes; (void)n_in; (void)out_size; (void)ws_size;
    const int*   yt  = (const int*)d_in[0];    // y_true int32 [512,16]
    const float* yp  = (const float*)d_in[1];  // y_pred f32   [512,16]
    float*       out = (float*)d_out;

    float* S = (float*)d_ws;                   // [512][512] = 1 MB
    float* U = S + (size_t)BN * BN;            // [16][512]  = 32 KB

    dim3 g1(BN / 16, BN / 16);                 // 32 x 32 blocks
    k_pairwise_hinge<<<g1, 256, 0, stream>>>(yp, S);
    k_mask_gemm<<<BN / 16, 32, 0, stream>>>(yt, S, U);
    k_finalize<<<1, 256, 0, stream>>>(yt, U, out);
}